// ChromaticityShiftSRGB_16853451669576
// MI455X (gfx1250) — compile-verified
//
#include <hip/hip_runtime.h>
#include <stdint.h>

// ---------------------------------------------------------------------------
// Vector types
// ---------------------------------------------------------------------------
typedef float        f32x4 __attribute__((ext_vector_type(4)));
typedef unsigned int u32x4 __attribute__((ext_vector_type(4)));
typedef int          i32x4 __attribute__((ext_vector_type(4)));
typedef int          i32x8 __attribute__((ext_vector_type(8)));

// ---------------------------------------------------------------------------
// Compile-time constants (exactly mirroring the reference math)
// ---------------------------------------------------------------------------
// Rec.709 RGB->XYZ (rows X,Y,Z), pre-scaled by 100 (reference multiplies X,Y,Z by 100)
constexpr double W00 = 0.4124564, W01 = 0.3575761, W02 = 0.1804375;
constexpr double W10 = 0.2126728, W11 = 0.7151522, W12 = 0.0721750;
constexpr double W20 = 0.0193339, W21 = 0.1191920, W22 = 0.9503041;

constexpr float A00 = (float)(100.0 * W00), A01 = (float)(100.0 * W01), A02 = (float)(100.0 * W02);
constexpr float A10 = (float)(100.0 * W10), A11 = (float)(100.0 * W11), A12 = (float)(100.0 * W12);
constexpr float A20 = (float)(100.0 * W20), A21 = (float)(100.0 * W21), A22 = (float)(100.0 * W22);

// inv(W) * 0.01  (reference: xyz2/100 then einsum with inv(W.T) == inv(W) @ xyz2)
constexpr double DET = W00*(W11*W22 - W12*W21) - W01*(W10*W22 - W12*W20) + W02*(W10*W21 - W11*W20);
constexpr float I00 = (float)(0.01 * ( (W11*W22 - W12*W21) / DET));
constexpr float I01 = (float)(0.01 * (-(W01*W22 - W02*W21) / DET));
constexpr float I02 = (float)(0.01 * ( (W01*W12 - W02*W11) / DET));
constexpr float I10 = (float)(0.01 * (-(W10*W22 - W12*W20) / DET));
constexpr float I11 = (float)(0.01 * ( (W00*W22 - W02*W20) / DET));
constexpr float I12 = (float)(0.01 * (-(W00*W12 - W02*W10) / DET));
constexpr float I20 = (float)(0.01 * ( (W10*W21 - W11*W20) / DET));
constexpr float I21 = (float)(0.01 * (-(W00*W21 - W01*W20) / DET));
constexpr float I22 = (float)(0.01 * ( (W00*W11 - W01*W10) / DET));

// D65 white point u', v'
constexpr double DEN65 = -2.0*0.31271 + 12.0*0.32902 + 3.0;
constexpr float  UNf   = (float)(4.0*0.31271 / DEN65);
constexpr float  VNf   = (float)(9.0*0.32902 / DEN65);

constexpr float T_CUT = (float)((6.0/29.0)*(6.0/29.0)*(6.0/29.0));      // 0.008856...
constexpr float L_LO  = (float)((29.0/3.0)*(29.0/3.0)*(29.0/3.0));      // 903.296...
constexpr float Y_LO  = (float)(100.0*(3.0/29.0)*(3.0/29.0)*(3.0/29.0));// 0.110705...

// Geometry (reference fixes [16, 3, 1024, 1024])
constexpr unsigned PLANE      = 1u << 20;     // pixels per plane (1024*1024)
constexpr unsigned TILE_PIX   = 1024;         // pixels per tile (TDM tile x-dim)
constexpr unsigned BLOCK_THREADS = 256;       // 8 wave32

// ---------------------------------------------------------------------------
// CDNA5 TDM availability (device pass only; clang-23 6-arg form)
// ---------------------------------------------------------------------------
#if defined(__gfx1250__) && defined(__has_builtin)
#  if __has_builtin(__builtin_amdgcn_tensor_load_to_lds) && __has_builtin(__builtin_amdgcn_s_wait_tensorcnt)
#    define HAS_TDM 1
#  endif
#endif
#ifndef HAS_TDM
#  define HAS_TDM 0
#endif

// ---------------------------------------------------------------------------
// Fast math primitives (native gfx1250 trans ops)
// ---------------------------------------------------------------------------
__device__ __forceinline__ float flog2(float x) { return __builtin_amdgcn_logf(x); }
__device__ __forceinline__ float fexp2(float x) { return __builtin_amdgcn_exp2f(x); }
__device__ __forceinline__ float frcp (float x) { return __builtin_amdgcn_rcpf(x); }

__device__ __forceinline__ float srgb2lin(float c) {
    float hi = fexp2(2.4f * flog2((c + 0.055f) * (1.0f / 1.055f)));
    return (c <= 0.04045f) ? c * (1.0f / 12.92f) : hi;
}

__device__ __forceinline__ float lin2srgb(float x) {
    x = fminf(fmaxf(x, 0.0f), 1.0f);
    float hi = 1.055f * fexp2((1.0f / 2.4f) * flog2(x)) - 0.055f;
    return (x <= 0.0031308f) ? x * 12.92f : hi;
}

// Full per-pixel chain: sRGB -> XYZ -> LUV(+5,-5) -> XYZ -> sRGB
__device__ __forceinline__ void xform(float r, float g, float b,
                                      float& ro, float& go, float& bo) {
    r = srgb2lin(r); g = srgb2lin(g); b = srgb2lin(b);

    float X = A00 * r + A01 * g + A02 * b;
    float Y = A10 * r + A11 * g + A12 * b;
    float Z = A20 * r + A21 * g + A22 * b;

    float rden = frcp(X + 15.0f * Y + 3.0f * Z);
    float u_p  = 4.0f * X * rden;
    float v_p  = 9.0f * Y * rden;

    float t      = Y * 0.01f;
    float cbrt_t = fexp2((1.0f / 3.0f) * flog2(t));
    float L      = (t <= T_CUT) ? L_LO * t : 116.0f * cbrt_t - 16.0f;

    float L13 = 13.0f * L;
    float u_s = L13 * (u_p - UNf) + 5.0f;
    float v_s = L13 * (v_p - VNf) - 5.0f;

    float r13 = frcp(L13);
    float u2  = u_s * r13 + UNf;
    float v2  = v_s * r13 + VNf;

    float a  = (L + 16.0f) * (1.0f / 116.0f);
    float Y2 = (L <= 8.0f) ? L * Y_LO : 100.0f * a * a * a;

    float r4v = frcp(4.0f * v2);
    float X2  = Y2 * 9.0f * u2 * r4v;
    float Z2  = Y2 * (12.0f - 3.0f * u2 - 20.0f * v2) * r4v;
    X2 = (X2 != X2) ? 0.0f : X2;   // nan_to_num(nan=0)
    Z2 = (Z2 != Z2) ? 0.0f : Z2;

    ro = lin2srgb(I00 * X2 + I01 * Y2 + I02 * Z2);
    go = lin2srgb(I10 * X2 + I11 * Y2 + I12 * Z2);
    bo = lin2srgb(I20 * X2 + I21 * Y2 + I22 * Z2);
}

#if HAS_TDM
// ---------------------------------------------------------------------------
// Issue one TDM 2D tile load: 1024 contiguous pixels x 3 planes -> LDS.
// D# per CDNA5 ISA (group0 + group1; groups 2/3 zero).
// clang-23 6-arg builtin: (g0 v4u32, g1 v8i32, g2 v4i32, g3 v4i32, v8i32, cpol)
// ---------------------------------------------------------------------------
__device__ __forceinline__ void tdm_issue(const float* im, int tile, const float* ldsDst) {
    unsigned long long q0  = (unsigned long long)tile << 10;        // first pixel of tile
    unsigned long long n   = q0 >> 20;                              // image index
    unsigned long long pix = q0 & (unsigned long long)(PLANE - 1);
    unsigned long long e0  = n * (3ull * PLANE) + pix;              // element offset of R
    unsigned long long ga  = (unsigned long long)(uintptr_t)im + (e0 << 2);
    unsigned           lad = (unsigned)(uintptr_t)ldsDst;           // low32 of flat LDS addr

    u32x4 g0;
    g0[0] = 1u;                                                     // count=1, no gather
    g0[1] = lad;                                                    // lds_addr
    g0[2] = (unsigned)(ga & 0xFFFFFFFFull);                         // global_addr[31:0]
    g0[3] = (unsigned)((ga >> 32) & 0x01FFFFFFull) | (2u << 30);    // addr[56:32] | type=2

    i32x8 g1;
    g1[0] = (int)(2u << 16);          // wg_mask=0, data_size=2 (4B), no pad/iterate
    g1[1] = 0;                        // barrier addr=0, tensor_dim0[15:0]=0
    g1[2] = (int)((3u << 16) | 0x4000u); // tensor_dim0=1<<30 (hi16), tensor_dim1=3
    g1[3] = (int)(TILE_PIX << 16);    // tensor_dim1 hi=0, tile_dim0=1024
    g1[4] = 3;                        // tile_dim1=3, tile_dim2=0
    g1[5] = (int)PLANE;               // tensor_dim0_stride = 1M elements
    g1[6] = 0;                        // stride0 hi, stride1 lo
    g1[7] = 0;                        // stride1 hi

    i32x4 gz4 = {0, 0, 0, 0};
    i32x8 gz8 = {0, 0, 0, 0, 0, 0, 0, 0};
    __builtin_amdgcn_tensor_load_to_lds(g0, g1, gz4, gz4, gz8, 0);
}
#endif

// ---------------------------------------------------------------------------
// Kernel: persistent blocks, each streams tiles of 1024 pixels (double-buffered
// TDM into LDS), transforms 4 pixels/lane, non-temporal b128 stores.
// ---------------------------------------------------------------------------
__global__ __launch_bounds__(BLOCK_THREADS) void luv_shift_kernel(
    const float* __restrict__ im, float* __restrict__ out, int nTiles, int gridB)
{
    const int tid   = threadIdx.x;
    const int tid4  = tid * 4;

#if HAS_TDM
    __shared__ __align__(16) float tileBuf[2][3][TILE_PIX];
    const bool wave0 = (tid >> 5) == 0;
    if (wave0) tdm_issue(im, (int)blockIdx.x, &tileBuf[0][0][0]);
#endif

    int p = 0;
    for (int t = (int)blockIdx.x; t < nTiles; t += gridB, p ^= 1) {
        // ---- stage inputs ----
#if HAS_TDM
        if (wave0) {
            int tn = t + gridB;
            if (tn < nTiles) {
                tdm_issue(im, tn, &tileBuf[p ^ 1][0][0]);
                __builtin_amdgcn_s_wait_tensorcnt(1);   // tile t complete, t+gridB in flight
            } else {
                __builtin_amdgcn_s_wait_tensorcnt(0);
            }
        }
        __syncthreads();                                // publish LDS tile to all 8 waves
        f32x4 r4 = *(const f32x4*)&tileBuf[p][0][tid4];
        f32x4 g4 = *(const f32x4*)&tileBuf[p][1][tid4];
        f32x4 b4 = *(const f32x4*)&tileBuf[p][2][tid4];
#endif

        // per-thread pixel/global offsets
        long long q    = (long long)t * TILE_PIX + tid4;        // global pixel index
        long long n    = q >> 20;
        long long pix  = q & (long long)(PLANE - 1);
        size_t    base = (size_t)n * (3u * PLANE) + (size_t)pix;

#if !HAS_TDM
        f32x4 r4 = __builtin_nontemporal_load((const f32x4*)(im + base));
        f32x4 g4 = __builtin_nontemporal_load((const f32x4*)(im + base + PLANE));
        f32x4 b4 = __builtin_nontemporal_load((const f32x4*)(im + base + 2u * PLANE));
#endif

        // ---- transform 4 pixels ----
        f32x4 ro, go, bo;
#pragma unroll
        for (int j = 0; j < 4; ++j) {
            float x, y, z;
            xform(r4[j], g4[j], b4[j], x, y, z);
            ro[j] = x; go[j] = y; bo[j] = z;
        }

        // ---- streaming stores (non-temporal: output is write-once) ----
        __builtin_nontemporal_store(ro, (f32x4*)(out + base));
        __builtin_nontemporal_store(go, (f32x4*)(out + base + PLANE));
        __builtin_nontemporal_store(bo, (f32x4*)(out + base + 2u * PLANE));

#if HAS_TDM
        __syncthreads();   // all waves done reading buf[p] before TDM reuses it
#endif
    }
}

// ---------------------------------------------------------------------------
// Launch
// ---------------------------------------------------------------------------
extern "C" void kernel_launch(void* const* d_in, const int* in_sizes, int n_in,
                              void* d_out, int out_size, void* d_ws, size_t ws_size,
                              hipStream_t stream) {
    const float* im  = (const float*)d_in[0];
    float*       out = (float*)d_out;

    long long total  = (long long)in_sizes[0];          // N*3*H*W
    int       nTiles = (int)(total / 3 / TILE_PIX);     // pixels / 1024
    int       gridB  = nTiles < 2048 ? nTiles : 2048;   // persistent-ish blocks

    hipLaunchKernelGGL(luv_shift_kernel, dim3(gridB), dim3(BLOCK_THREADS), 0, stream,
                       im, out, nTiles, gridB);
}